// gather_module_32933809226253
// MI455X (gfx1250) — compile-verified
//
#include <hip/hip_runtime.h>

typedef float v2f __attribute__((ext_vector_type(2)));
typedef float v8f __attribute__((ext_vector_type(8)));

#define HD 512
#define WD 512
#define HO 1024
#define WO 1024
#define NB 16

// One output pixel per lane. 32 consecutive ox per wave.
// Per-pixel 3x3 channel matmul done with V_WMMA_F32_16X16X4_F32 (2 WMMAs / wave).
__global__ __launch_bounds__(256) void fuse_upconv_kernel(
    const float* __restrict__ mask,   // (16,2,512,512)
    const float* __restrict__ xl,     // (16,3,1024,1024)
    const float* __restrict__ yl,     // (16,3,1024,1024)
    const float* __restrict__ cw,     // (3,3) [o][i]
    const float* __restrict__ cb,     // (3,)
    float* __restrict__ out,          // (16,3,1024,1024)
    float* __restrict__ xmo,          // (16,1024,1024)
    float* __restrict__ ymo)          // (16,1024,1024)
{
    const int t    = blockIdx.x * blockDim.x + threadIdx.x;
    const int lane = threadIdx.x & 31;
    const int ox = t & (WO - 1);
    const int oy = (t >> 10) & (HO - 1);
    const int b  = t >> 20;

    // ---- bilinear 2x upsample, half-pixel centers: src = o/2 - 0.25 ----
    const int py = oy & 1, px = ox & 1;
    const int jy = oy >> 1, jx = ox >> 1;
    const int y0 = py ? jy : (jy > 0 ? jy - 1 : 0);
    const int y1 = py ? (jy < HD - 1 ? jy + 1 : HD - 1) : jy;
    const int x0 = px ? jx : (jx > 0 ? jx - 1 : 0);
    const int x1 = px ? (jx < WD - 1 ? jx + 1 : WD - 1) : jx;
    const float wyA = py ? 0.75f : 0.25f, wyB = py ? 0.25f : 0.75f;
    const float wxA = px ? 0.75f : 0.25f, wxB = px ? 0.25f : 0.75f;

    const float* m0 = mask + ((size_t)b * 2 + 0) * (size_t)(HD * WD);
    const float* m1 = m0 + (size_t)(HD * WD);
    const int r0 = y0 * WD, r1 = y1 * WD;
    // mask is reused ~4x and fits in L2 (33.5 MB << 192 MB): regular-temporal loads
    const float xm = wyA * (wxA * m0[r0 + x0] + wxB * m0[r0 + x1])
                   + wyB * (wxA * m0[r1 + x0] + wxB * m0[r1 + x1]);
    const float ym = wyA * (wxA * m1[r0 + x0] + wxB * m1[r0 + x1])
                   + wyB * (wxA * m1[r1 + x0] + wxB * m1[r1 + x1]);

    // ---- fused = xm*xl + ym*yl  (streaming: non-temporal) ----
    const size_t plane = (size_t)HO * WO;
    const size_t pix   = (size_t)oy * WO + ox;
    const float* xb = xl + (size_t)b * 3 * plane + pix;
    const float* yb = yl + (size_t)b * 3 * plane + pix;
    const float f0 = xm * __builtin_nontemporal_load(xb)
                   + ym * __builtin_nontemporal_load(yb);
    const float f1 = xm * __builtin_nontemporal_load(xb + plane)
                   + ym * __builtin_nontemporal_load(yb + plane);
    const float f2 = xm * __builtin_nontemporal_load(xb + 2 * plane)
                   + ym * __builtin_nontemporal_load(yb + 2 * plane);

    // ---- per-pixel 3x3 channel matmul via V_WMMA_F32_16X16X4_F32 ----
    // A (16x4): A[m][k] = cw[m*3+k] for m<3,k<3 else 0.
    //   layout: lanes 0-15 M=0..15; VGPR0 = K0|K2 (low|high half), VGPR1 = K1|K3.
    const int  mrow = lane & 15;
    const bool lo   = lane < 16;
    const bool mv   = mrow < 3;
    v2f A;
    A.x = mv ? (lo ? cw[mrow * 3 + 0] : cw[mrow * 3 + 2]) : 0.0f;
    A.y = (mv && lo) ? cw[mrow * 3 + 1] : 0.0f;

    // C (16x16): bias on rows M=0..2. VGPR r, lanes<16 -> M=r; lanes>=16 -> M=r+8.
    v8f C = {};
    C[0] = lo ? cb[0] : 0.0f;
    C[1] = lo ? cb[1] : 0.0f;
    C[2] = lo ? cb[2] : 0.0f;

    // B (4x16): VGPR0 = K0|K2, VGPR1 = K1|K3 split across lane halves; N = lane%16.
    // Wave holds fused for 32 pixels (pixel == lane); B0 covers pixels 0-15, B1 16-31.
    const int src = lane & 15;
    const float t0 = __shfl(f0, src, 32);
    const float t1 = __shfl(f1, src, 32);
    const float t2 = __shfl(f2, src, 32);
    v2f B0; B0.x = lo ? t0 : t2; B0.y = lo ? t1 : 0.0f;
    const float u0 = __shfl(f0, src + 16, 32);
    const float u1 = __shfl(f1, src + 16, 32);
    const float u2 = __shfl(f2, src + 16, 32);
    v2f B1; B1.x = lo ? u0 : u2; B1.y = lo ? u1 : 0.0f;

    const v8f D0 = __builtin_amdgcn_wmma_f32_16x16x4_f32(
        false, A, false, B0, (short)0, C, false, false);
    const v8f D1 = __builtin_amdgcn_wmma_f32_16x16x4_f32(
        false, A, false, B1, (short)0, C, false, false);

    // ---- ReLU + coalesced NT stores: lane l stores its own pixel ----
    float* ob = out + (size_t)b * 3 * plane + pix;
#pragma unroll
    for (int o = 0; o < 3; ++o) {
        const float dhi = __shfl(D1[o], src, 32);      // pixel 16+n lives in lane n of D1
        float v = lo ? D0[o] : dhi;
        v = fmaxf(v, 0.0f);
        __builtin_nontemporal_store(v, ob + (size_t)o * plane);
    }
    __builtin_nontemporal_store(xm, xmo + (size_t)b * plane + pix);
    __builtin_nontemporal_store(ym, ymo + (size_t)b * plane + pix);
}

extern "C" void kernel_launch(void* const* d_in, const int* in_sizes, int n_in,
                              void* d_out, int out_size, void* d_ws, size_t ws_size,
                              hipStream_t stream) {
    const float* mask = (const float*)d_in[0];
    const float* xl   = (const float*)d_in[1];
    const float* yl   = (const float*)d_in[2];
    const float* cw   = (const float*)d_in[3];
    const float* cb   = (const float*)d_in[4];

    float* out = (float*)d_out;                               // (16,3,1024,1024)
    float* xmo = out + (size_t)NB * 3 * HO * WO;              // (16,1,1024,1024)
    float* ymo = xmo + (size_t)NB * HO * WO;                  // (16,1,1024,1024)

    const int total = NB * HO * WO;                           // 2^24, exact multiple of 256
    hipLaunchKernelGGL(fuse_upconv_kernel,
                       dim3(total / 256), dim3(256), 0, stream,
                       mask, xl, yl, cw, cb, out, xmo, ymo);
}